// GCN_84576495993582
// MI455X (gfx1250) — compile-verified
//
#include <hip/hip_runtime.h>

// ---------------------------------------------------------------------------
// GCN forward (2x GCNConv + FC) for MI455X / gfx1250.
// GEMMs use V_WMMA_F32_16X16X4_F32 (fp32 matrix pipe -> numerically faithful
// to the fp32 reference). Each wave computes a 16x64 output strip: the A
// fragment is loaded once per K-step and feeds 4 independent WMMA accumulator
// chains (ILP across the XDL pipe). Edge propagation uses fp32 hardware
// atomics; the 102 MB feature matrix is L2-resident (192 MB L2), so the
// gather/scatter phase is the bound, not FLOPs.
// ---------------------------------------------------------------------------

typedef __attribute__((ext_vector_type(2))) float v2f;
typedef __attribute__((ext_vector_type(8))) float v8f;

#define NNODES 100000
#define NEDGES 1600000

static __device__ __forceinline__ void atomic_add_f32(float* p, float v) {
  __hip_atomic_fetch_add(p, v, __ATOMIC_RELAXED, __HIP_MEMORY_SCOPE_AGENT);
}

// ---- small elementwise kernels ---------------------------------------------

__global__ void k_fill(float* __restrict__ p, float v, int n) {
  int i = blockIdx.x * blockDim.x + threadIdx.x;
  if (i < n) p[i] = v;
}

__global__ void k_deg_scatter(const int* __restrict__ col, const float* __restrict__ ew,
                              float* __restrict__ deg, int E) {
  int e = blockIdx.x * blockDim.x + threadIdx.x;
  if (e < E) atomic_add_f32(&deg[col[e]], ew[e]);
}

__global__ void k_rsqrt_inplace(float* __restrict__ d, int n) {
  int i = blockIdx.x * blockDim.x + threadIdx.x;
  if (i < n) {
    float v = d[i];
    d[i] = v > 0.f ? rsqrtf(v) : 0.f;
  }
}

__global__ void k_edge_norm(const int* __restrict__ row, const int* __restrict__ col,
                            const float* __restrict__ ew, const float* __restrict__ dinv,
                            float* __restrict__ norm, int E) {
  int e = blockIdx.x * blockDim.x + threadIdx.x;
  if (e < E) norm[e] = dinv[row[e]] * ew[e] * dinv[col[e]];
}

// out[i][:] = dinv[i]^2 * h[i][:]   (self-loop term: norm = dinv*1*dinv)
__global__ void k_self_init(const float* __restrict__ h, const float* __restrict__ dinv,
                            float* __restrict__ out, int n4, int Fd4) {
  int idx = blockIdx.x * blockDim.x + threadIdx.x;
  if (idx >= n4) return;
  int i = idx / Fd4;
  float d = dinv[i];
  float s = d * d;
  float4 v = ((const float4*)h)[idx];
  float4 o;
  o.x = v.x * s; o.y = v.y * s; o.z = v.z * s; o.w = v.w * s;
  ((float4*)out)[idx] = o;
}

// out[col[e]][:] += norm[e] * h[row[e]][:]   (float4 per thread, fp32 atomics)
__global__ void k_edge_scatter(const int* __restrict__ row, const int* __restrict__ col,
                               const float* __restrict__ norm, const float* __restrict__ h,
                               float* __restrict__ out, int E, int F) {
  int idx = blockIdx.x * blockDim.x + threadIdx.x;
  int Fd4 = F >> 2;
  int e = idx / Fd4;
  if (e >= E) return;
  int c = idx - e * Fd4;
  float w = norm[e];
  int r = row[e];
  int cl = col[e];
  float4 v = *((const float4*)(h + (size_t)r * F) + c);
  float* o = out + (size_t)cl * F + (c << 2);
  atomic_add_f32(o + 0, v.x * w);
  atomic_add_f32(o + 1, v.y * w);
  atomic_add_f32(o + 2, v.z * w);
  atomic_add_f32(o + 3, v.w * w);
}

__global__ void k_bias_relu(float* __restrict__ h, const float* __restrict__ bias,
                            int n4, int Fd4) {
  int idx = blockIdx.x * blockDim.x + threadIdx.x;
  if (idx >= n4) return;
  int c = idx % Fd4;
  float4 v = ((float4*)h)[idx];
  float4 b = ((const float4*)bias)[c];
  v.x = fmaxf(v.x + b.x, 0.f);
  v.y = fmaxf(v.y + b.y, 0.f);
  v.z = fmaxf(v.z + b.z, 0.f);
  v.w = fmaxf(v.w + b.w, 0.f);
  ((float4*)h)[idx] = v;
}

// ---- fp32 WMMA GEMM, wide path: C(MxF) = A(MxK) @ B(KxF) -------------------
//
// V_WMMA_F32_16X16X4_F32 fragment layout (CDNA5 ISA):
//   A 16x4 : m = lane%16, k = k0 + (lane/16)*2 + {0,1}  -> contiguous f32 pair
//   B 4x16 : n = lane%16, same k pairing
//   C/D    : lanes 0-15 -> M = r (VGPR r), lanes 16-31 -> M = 8+r; N = lane%16
//
// One wave computes a 16 x (NT*16) strip: A fragment loaded once per K-step,
// NT independent accumulator chains (WMMA ILP). Requires F % (blockDim.y*NT*16)
// == 0 and M % 16 == 0 (100000 = 6250*16). No masking in the hot loop.
template <int NT>
__global__ void k_gemm_wmma_wide(const float* __restrict__ A, const float* __restrict__ B,
                                 float* __restrict__ C, int M, int K, int F) {
  const int lane   = threadIdx.x;                        // 0..31
  const int tileM  = blockIdx.x;
  const int tileN0 = (blockIdx.y * blockDim.y + threadIdx.y) * NT;  // 16-col units
  const int mloc   = lane & 15;
  const int koff   = (lane >> 4) << 1;                   // 0 or 2
  const int m      = tileM * 16 + mloc;

  v8f acc[NT] = {};

  const float* arow = A + (size_t)m * K;
  for (int k = 0; k < K; k += 4) {
    v2f a;
    a.x = arow[k + koff];
    a.y = arow[k + koff + 1];
    const float* b0 = B + (size_t)(k + koff) * F + tileN0 * 16 + mloc;
    #pragma unroll
    for (int s = 0; s < NT; ++s) {
      v2f b;
      b.x = b0[s * 16];        // B[k+koff  ][n_s]
      b.y = b0[s * 16 + F];    // B[k+koff+1][n_s]
      acc[s] = __builtin_amdgcn_wmma_f32_16x16x4_f32(false, a, false, b,
                                                     (short)0, acc[s], false, false);
    }
  }

  const int mbase = (lane < 16) ? 0 : 8;
  #pragma unroll
  for (int s = 0; s < NT; ++s) {
    const int n = (tileN0 + s) * 16 + mloc;
    #pragma unroll
    for (int r = 0; r < 8; ++r) {
      C[(size_t)(tileM * 16 + mbase + r) * F + n] = acc[s][r];
    }
  }
}

// ---- fp32 WMMA GEMM, ragged path (FC head, F=40): clamp+mask B, guard store
__global__ void k_gemm_wmma_ragged(const float* __restrict__ A, const float* __restrict__ B,
                                   const float* __restrict__ bias, float* __restrict__ C,
                                   int M, int K, int F) {
  const int lane  = threadIdx.x;
  const int tileM = blockIdx.x;
  const int tileN = blockIdx.y * blockDim.y + threadIdx.y;
  const int mloc  = lane & 15;
  const int koff  = (lane >> 4) << 1;
  const int m     = tileM * 16 + mloc;
  const int n     = tileN * 16 + mloc;
  const int nc    = n < F ? n : F - 1;      // clamped (safe) address
  const float nmask = n < F ? 1.f : 0.f;

  v8f acc = {};
  const float* arow = A + (size_t)m * K;
  for (int k = 0; k < K; k += 4) {
    v2f a;
    a.x = arow[k + koff];
    a.y = arow[k + koff + 1];
    v2f b;
    b.x = B[(size_t)(k + koff) * F + nc] * nmask;
    b.y = B[(size_t)(k + koff + 1) * F + nc] * nmask;
    acc = __builtin_amdgcn_wmma_f32_16x16x4_f32(false, a, false, b,
                                                (short)0, acc, false, false);
  }

  if (n < F) {
    const int mbase = (lane < 16) ? 0 : 8;
    const float bv = bias ? bias[n] : 0.f;
    #pragma unroll
    for (int r = 0; r < 8; ++r) {
      C[(size_t)(tileM * 16 + mbase + r) * F + n] = acc[r] + bv;
    }
  }
}

// ---------------------------------------------------------------------------

extern "C" void kernel_launch(void* const* d_in, const int* in_sizes, int n_in,
                              void* d_out, int out_size, void* d_ws, size_t ws_size,
                              hipStream_t stream) {
  const float* x   = (const float*)d_in[0];
  const int*   ei  = (const int*)d_in[1];   // [2, E]: row then col
  const float* ea  = (const float*)d_in[2];
  const float* W1  = (const float*)d_in[3];
  const float* b1  = (const float*)d_in[4];
  const float* W2  = (const float*)d_in[5];
  const float* b2  = (const float*)d_in[6];
  const float* fcW = (const float*)d_in[7];
  const float* fcb = (const float*)d_in[8];
  float* out = (float*)d_out;

  const int* row = ei;
  const int* col = ei + NEDGES;

  // workspace layout (floats): deg | norm | bufA | bufB  (~212 MB total)
  float* deg  = (float*)d_ws;                       // NNODES (reused as dinv)
  float* norm = deg  + 102400;                      // NEDGES
  float* bufA = norm + NEDGES;                      // NNODES*256
  float* bufB = bufA + (size_t)NNODES * 256;        // NNODES*256

  const int TB = 256;
  dim3 b256(TB);
  dim3 gblk(32, 4);                                  // 4 waves / block
  const int n4 = NNODES * 64;                        // N*256/4 float4 slots
  const int e4 = NEDGES * 64;                        // E*256/4 float4 slots

  // --- normalization coefficients (graph-only, shared by both layers) ---
  k_fill<<<(NNODES + TB - 1) / TB, b256, 0, stream>>>(deg, 1.0f, NNODES); // self loop
  k_deg_scatter<<<(NEDGES + TB - 1) / TB, b256, 0, stream>>>(col, ea, deg, NEDGES);
  k_rsqrt_inplace<<<(NNODES + TB - 1) / TB, b256, 0, stream>>>(deg, NNODES);
  k_edge_norm<<<(NEDGES + TB - 1) / TB, b256, 0, stream>>>(row, col, ea, deg, norm, NEDGES);

  // --- layer 1: bufB = relu(Ahat (x @ W1) + b1) ---
  // one block (4 waves x 16x64 strips) covers the full 256-wide output row band
  k_gemm_wmma_wide<4><<<dim3(NNODES / 16, 1), gblk, 0, stream>>>(x, W1, bufA,
                                                                 NNODES, 128, 256);
  k_self_init<<<(n4 + TB - 1) / TB, b256, 0, stream>>>(bufA, deg, bufB, n4, 64);
  k_edge_scatter<<<(e4 + TB - 1) / TB, b256, 0, stream>>>(row, col, norm, bufA, bufB,
                                                          NEDGES, 256);
  k_bias_relu<<<(n4 + TB - 1) / TB, b256, 0, stream>>>(bufB, b1, n4, 64);

  // --- layer 2: bufB = relu(Ahat (bufB @ W2) + b2) ---
  k_gemm_wmma_wide<4><<<dim3(NNODES / 16, 1), gblk, 0, stream>>>(bufB, W2, bufA,
                                                                 NNODES, 256, 256);
  k_self_init<<<(n4 + TB - 1) / TB, b256, 0, stream>>>(bufA, deg, bufB, n4, 64);
  k_edge_scatter<<<(e4 + TB - 1) / TB, b256, 0, stream>>>(row, col, norm, bufA, bufB,
                                                          NEDGES, 256);
  k_bias_relu<<<(n4 + TB - 1) / TB, b256, 0, stream>>>(bufB, b2, n4, 64);

  // --- FC head: out = bufB @ fcW + fcb   (F=40: 3 ragged tiles, 4th masked) ---
  k_gemm_wmma_ragged<<<dim3(NNODES / 16, 1), gblk, 0, stream>>>(bufB, fcW, fcb, out,
                                                                NNODES, 256, 40);
}